// CRF_41231686041799
// MI455X (gfx1250) — compile-verified
//
#include <hip/hip_runtime.h>

#define NB     8192
#define LL     512
#define TT     6
#define TSTART 4
#define TSTOP  5
#define NEGV   (-10000.0f)
#define CH     8
#define LCH    (LL / CH)   // 64 chunks

typedef __attribute__((ext_vector_type(2))) float v2f;
typedef __attribute__((ext_vector_type(8))) float v8f;

// Load one 8-step chunk: 192B of feats as 12x b128, 8 int64 tags as 4x b128.
__device__ __forceinline__ void load_chunk(const float* __restrict__ fb,
                                           const int* __restrict__ tb,
                                           int pc, float (&F)[CH * 6], int (&TG)[CH]) {
    const float4* p4 = (const float4*)(fb + pc * (CH * 6));   // 16B-aligned
    #pragma unroll
    for (int q = 0; q < (CH * 6) / 4; ++q) {
        float4 v = p4[q];
        F[4 * q + 0] = v.x; F[4 * q + 1] = v.y;
        F[4 * q + 2] = v.z; F[4 * q + 3] = v.w;
    }
    const int4* t4 = (const int4*)(tb + pc * (CH * 2));       // 16B-aligned
    #pragma unroll
    for (int q = 0; q < CH / 2; ++q) {
        int4 w = t4[q];
        TG[2 * q + 0] = w.x;   // low dword of int64 tag (values 0..3)
        TG[2 * q + 1] = w.z;
    }
}

__device__ __forceinline__ void compute_chunk(int c, int len,
        const float (&F)[CH * 6], const int (&TG)[CH],
        float (&a)[6], const float (&W)[36], const float* sm,
        float &gold, int &prev) {
    #pragma unroll
    for (int k = 0; k < CH; ++k) {
        int t = c * CH + k;
        bool act = t < len;
        // forward recurrence: a' = log(exp(a-m) . W^T) + m + feat
        float m = fmaxf(fmaxf(fmaxf(a[0], a[1]), fmaxf(a[2], a[3])), fmaxf(a[4], a[5]));
        float e[6];
        #pragma unroll
        for (int i = 0; i < 6; ++i) e[i] = __expf(a[i] - m);
        float na[6];
        #pragma unroll
        for (int j = 0; j < 6; ++j) {
            float s = e[0] * W[j * 6];
            #pragma unroll
            for (int i = 1; i < 6; ++i) s = __builtin_fmaf(e[i], W[j * 6 + i], s);
            na[j] = __logf(s) + m + F[k * 6 + j];
        }
        #pragma unroll
        for (int j = 0; j < 6; ++j) a[j] = act ? na[j] : a[j];
        // gold path score
        int tg = TG[k];
        float emit = F[k * 6 + 0];
        emit = (tg == 1) ? F[k * 6 + 1] : emit;
        emit = (tg == 2) ? F[k * 6 + 2] : emit;
        emit = (tg == 3) ? F[k * 6 + 3] : emit;
        float tsc = sm[tg * 6 + prev];
        gold = act ? (gold + tsc + emit) : gold;
        prev = act ? tg : prev;
    }
}

__global__ __launch_bounds__(64) void crf_fwd_kernel(
        const float* __restrict__ feats, const float* __restrict__ trans,
        const int* __restrict__ tags2, const int* __restrict__ lens,
        float* __restrict__ part)
{
    feats = (const float*)__builtin_assume_aligned(feats, 16);
    tags2 = (const int*)__builtin_assume_aligned(tags2, 16);

    __shared__ float sm[36];
    int tid = threadIdx.x;
    if (tid < 36) sm[tid] = trans[tid];
    __syncthreads();

    int b = blockIdx.x * 64 + tid;
    int len = lens[b];                       // 1..512

    // W = exp(trans), uniform per wave (scalar loads feed VALU exps)
    float W[36];
    #pragma unroll
    for (int k = 0; k < 36; ++k) W[k] = __expf(trans[k]);

    float a[6];
    #pragma unroll
    for (int i = 0; i < 6; ++i) a[i] = NEGV;
    a[TSTART] = 0.0f;

    const float* fb = feats + (size_t)b * (LL * TT);   // 12288B stride: 16B-aligned
    const int*   tb = tags2 + (size_t)b * (LL * 2);    // 4096B stride: 16B-aligned

    float gold = 0.0f;
    int prev = TSTART;

    float F0[CH * 6], F1[CH * 6];
    int   T0[CH],     T1[CH];
    load_chunk(fb, tb, 0, F0, T0);

    int nch = (len + CH - 1) / CH;
    int c = 0;
    while (true) {
        {
            int pc = (c + 1 < LCH) ? (c + 1) : (LCH - 1);
            load_chunk(fb, tb, pc, F1, T1);          // double-buffered prefetch
            if (c + 2 < LCH) {
                const char* pp = (const char*)(fb + (size_t)(c + 2) * CH * 6);
                __builtin_prefetch(pp, 0, 0);        // global_prefetch_b8
                __builtin_prefetch(pp + 128, 0, 0);
            }
            compute_chunk(c, len, F0, T0, a, W, sm, gold, prev);
        }
        if (++c >= nch) break;
        {
            int pc = (c + 1 < LCH) ? (c + 1) : (LCH - 1);
            load_chunk(fb, tb, pc, F0, T0);
            if (c + 2 < LCH) {
                const char* pp = (const char*)(fb + (size_t)(c + 2) * CH * 6);
                __builtin_prefetch(pp, 0, 0);
                __builtin_prefetch(pp + 128, 0, 0);
            }
            compute_chunk(c, len, F1, T1, a, W, sm, gold, prev);
        }
        if (++c >= nch) break;
    }

    // terminal transition for gold
    int lt = tb[2 * (len - 1)];
    gold += sm[TSTOP * 6 + lt];

    // log Z = logsumexp(a + trans[STOP, :])
    float z[6];
    #pragma unroll
    for (int j = 0; j < 6; ++j) z[j] = a[j] + trans[TSTOP * 6 + j];
    float m2 = fmaxf(fmaxf(fmaxf(z[0], z[1]), fmaxf(z[2], z[3])), fmaxf(z[4], z[5]));
    float s2 = 0.0f;
    #pragma unroll
    for (int j = 0; j < 6; ++j) s2 += __expf(z[j] - m2);
    float lz = __logf(s2) + m2;

    part[b] = lz - gold;
}

// Single-wave reduction of 8192 partials via chained f32 WMMA column sums.
// A = all-ones 16x4, B = 64 data values -> D accumulates column sums.
__global__ __launch_bounds__(32) void crf_reduce_kernel(const float* __restrict__ part,
                                                        float* __restrict__ out)
{
    int lane = threadIdx.x;                  // 0..31, all active (EXEC all-ones)
    v8f acc = {};
    v2f ones; ones.x = 1.0f; ones.y = 1.0f;
    int off = lane + (lane & 16);            // bijective B-matrix slot mapping
    for (int it = 0; it < NB / 64; ++it) {   // 128 chained WMMAs
        v2f bv;
        bv.x = part[it * 64 + off];
        bv.y = part[it * 64 + off + 16];
        acc = __builtin_amdgcn_wmma_f32_16x16x4_f32(
            /*neg_a=*/false, ones, /*neg_b=*/false, bv,
            /*c_mod=*/(short)0, acc, /*reuse_a=*/false, /*reuse_b=*/false);
    }
    float v = acc[0];                        // row 0 = column sums across lanes
    v += __shfl_xor(v, 1, 32);
    v += __shfl_xor(v, 2, 32);
    v += __shfl_xor(v, 4, 32);
    v += __shfl_xor(v, 8, 32);               // lanes 0..15 now hold the grand total
    if (lane == 0) out[0] = v * (1.0f / (float)NB);
}

extern "C" void kernel_launch(void* const* d_in, const int* in_sizes, int n_in,
                              void* d_out, int out_size, void* d_ws, size_t ws_size,
                              hipStream_t stream) {
    const float* feats = (const float*)d_in[0];
    const float* trans = (const float*)d_in[1];
    const int*   tags2 = (const int*)d_in[2];   // int64 tags viewed as int pairs
    const int*   lens  = (const int*)d_in[3];
    float* part = (float*)d_ws;                 // 8192 floats of scratch

    crf_fwd_kernel<<<NB / 64, 64, 0, stream>>>(feats, trans, tags2, lens, part);
    crf_reduce_kernel<<<1, 32, 0, stream>>>(part, (float*)d_out);
}